// QuantumSelfAttention_65481071407779
// MI455X (gfx1250) — compile-verified
//
#include <hip/hip_runtime.h>
#include <hip/hip_bf16.h>
#include <math.h>

typedef __attribute__((ext_vector_type(2))) float v2f;
typedef __attribute__((ext_vector_type(8))) float v8f;

// ds_swizzle SWAPX16: group-of-32 mode, xor_mask=0x10, or=0, and_mask=0x1f
#define SWZ_SWAPX16 0x401F
__device__ __forceinline__ float swap16(float v) {
    return __int_as_float(__builtin_amdgcn_ds_swizzle(__float_as_int(v), SWZ_SWAPX16));
}

// ---------------------------------------------------------------------------
// Kernel 1: build the fixed 32x4 real matrix V from the 15 circuit params.
// Lane j (0..3) simulates the fixed portion of the circuit applied to basis
// state e_{4j}, giving column j of the 16x4 complex matrix M.  V packs
// [Re;Im] with the -i factors of the RX encoding absorbed:
//   state_re = u0*Mr0 + u1*Mr3 + u2*(-Mi1) + u3*(-Mi2)
//   state_im = u0*Mi0 + u1*Mi3 + u2*( Mr1) + u3*( Mr2)
// with u = (c0c1, -s0s1, -c0s1, -s0c1).  V row-major: V[row*4+col], rows
// 0..15 = Re, 16..31 = Im.  One-time cost; precision kept with libm sincosf.
// ---------------------------------------------------------------------------
__global__ void qsa_precompute(const float* __restrict__ rot,
                               const float* __restrict__ ent,
                               float* __restrict__ V) {
    const int j = threadIdx.x;
    if (j >= 4) return;

    float re[16], im[16];
#pragma unroll
    for (int m = 0; m < 16; ++m) { re[m] = 0.f; im[m] = 0.f; }
    re[4 * j] = 1.0f;

#pragma unroll
    for (int w = 0; w < 4; ++w) {
        const int st = 8 >> w;   // wire w stride (wire 0 = MSB)
        // RX
        {
            float s, c; sincosf(0.5f * rot[3 * w], &s, &c);
#pragma unroll
            for (int m = 0; m < 16; ++m) if (!(m & st)) {
                const int a = m, b = m | st;
                const float r0 = re[a], i0 = im[a], r1 = re[b], i1 = im[b];
                re[a] = c * r0 + s * i1;  im[a] = c * i0 - s * r1;
                re[b] = c * r1 + s * i0;  im[b] = c * i1 - s * r0;
            }
        }
        // RY
        {
            float s, c; sincosf(0.5f * rot[3 * w + 1], &s, &c);
#pragma unroll
            for (int m = 0; m < 16; ++m) if (!(m & st)) {
                const int a = m, b = m | st;
                const float r0 = re[a], i0 = im[a], r1 = re[b], i1 = im[b];
                re[a] = c * r0 - s * r1;  im[a] = c * i0 - s * i1;
                re[b] = s * r0 + c * r1;  im[b] = s * i0 + c * i1;
            }
        }
        // RZ
        {
            float s, c; sincosf(0.5f * rot[3 * w + 2], &s, &c);
#pragma unroll
            for (int m = 0; m < 16; ++m) if (!(m & st)) {
                const int a = m, b = m | st;
                const float r0 = re[a], i0 = im[a], r1 = re[b], i1 = im[b];
                re[a] = c * r0 + s * i0;  im[a] = c * i0 - s * r0;
                re[b] = c * r1 - s * i1;  im[b] = c * i1 + s * r1;
            }
        }
    }

#pragma unroll
    for (int i = 0; i < 3; ++i) {
        const int sc = 8 >> i;
        const int stt = 8 >> (i + 1);
        // CNOT(i, i+1)
#pragma unroll
        for (int m = 0; m < 16; ++m) if ((m & sc) && !(m & stt)) {
            const int a = m, b = m | stt;
            float tr = re[a]; re[a] = re[b]; re[b] = tr;
            float ti = im[a]; im[a] = im[b]; im[b] = ti;
        }
        // RZ(ent[i]) on wire i
        {
            float s, c; sincosf(0.5f * ent[i], &s, &c);
#pragma unroll
            for (int m = 0; m < 16; ++m) if (!(m & sc)) {
                const int a = m, b = m | sc;
                const float r0 = re[a], i0 = im[a], r1 = re[b], i1 = im[b];
                re[a] = c * r0 + s * i0;  im[a] = c * i0 - s * r0;
                re[b] = c * r1 - s * i1;  im[b] = c * i1 + s * r1;
            }
        }
    }

    int col; bool realCoeff;
    if (j == 0)      { col = 0; realCoeff = true;  }
    else if (j == 3) { col = 1; realCoeff = true;  }
    else if (j == 1) { col = 2; realCoeff = false; }
    else             { col = 3; realCoeff = false; }

#pragma unroll
    for (int r = 0; r < 16; ++r) {
        if (realCoeff) {
            V[r * 4 + col]        = re[r];
            V[(16 + r) * 4 + col] = im[r];
        } else {
            V[r * 4 + col]        = -im[r];
            V[(16 + r) * 4 + col] = re[r];
        }
    }
}

// ---------------------------------------------------------------------------
// Kernel 2: each wave handles 32 samples per iteration (two 16-sample WMMA
// groups).  Each lane computes sincos for its OWN sample; a SWAPX16
// ds_swizzle exchange then supplies the partner lane-half's K-columns, so
// trig is computed exactly once per sample.
//
// A layout (32-bit A 16x4, ISA 7.12.2): lane L holds row M = L%16;
//   VGPR0 = K=0 (lanes 0-15) / K=2 (lanes 16-31); VGPR1 = K=1 / K=3.
// B layout (4x16): lanes 0-15 hold K=0..1, lanes 16-31 K=2..3; N = lane%16.
// D layout (16x16 f32): element (m = r + 8*(lane>=16), n = lane%16), VGPR r.
// ---------------------------------------------------------------------------
__global__ void qsa_main(const float2* __restrict__ x,
                         const float* __restrict__ V,
                         float* __restrict__ out,
                         int nblk, int nwaves) {
    const int tid  = blockIdx.x * blockDim.x + threadIdx.x;
    const int wave = tid >> 5;
    const int lane = threadIdx.x & 31;
    const int nloc = lane & 15;
    const bool lo  = lane < 16;
    const int kb   = lo ? 0 : 2;

    // Fixed-matrix WMMA A operands (loaded once per wave).
    v2f a0, a1;
    a0.x = V[nloc * 4 + kb];          a0.y = V[nloc * 4 + kb + 1];
    a1.x = V[(16 + nloc) * 4 + kb];   a1.y = V[(16 + nloc) * 4 + kb + 1];

    for (int g = wave; g < nblk; g += nwaves) {
        const int base = g * 32;              // 32 samples per iteration
        const int smp  = base + lane;         // this lane's own sample

        const float2 xv = x[smp];
        float s0, c0, s1, c1;
        __sincosf(0.5f * xv.x, &s0, &c0);
        __sincosf(0.5f * xv.y, &s1, &c1);
        const float u0 =  c0 * c1;
        const float u1 = -s0 * s1;
        const float u2 = -c0 * s1;
        const float u3 = -s0 * c1;

        // Partner lane-half's coefficients (sample lane^16).
        const float w0 = swap16(u0);
        const float w1 = swap16(u1);
        const float w2 = swap16(u2);
        const float w3 = swap16(u3);

        // Group 0 = samples base..base+15:  lanes<16 own (u0,u1);
        // lanes>=16 need partner's (u2,u3).
        v2f bA; bA.x = lo ? u0 : w2;  bA.y = lo ? u1 : w3;
        // Group 1 = samples base+16..base+31: lanes<16 need partner's
        // (u0,u1); lanes>=16 own (u2,u3).
        v2f bB; bB.x = lo ? w0 : u2;  bB.y = lo ? w1 : u3;

        v8f cz = {};
        v8f dA0 = __builtin_amdgcn_wmma_f32_16x16x4_f32(
            false, a0, false, bA, (short)0, cz, false, false);
        v8f dA1 = __builtin_amdgcn_wmma_f32_16x16x4_f32(
            false, a1, false, bA, (short)0, cz, false, false);
        v8f dB0 = __builtin_amdgcn_wmma_f32_16x16x4_f32(
            false, a0, false, bB, (short)0, cz, false, false);
        v8f dB1 = __builtin_amdgcn_wmma_f32_16x16x4_f32(
            false, a1, false, bB, (short)0, cz, false, false);

        // Per-group: probabilities + Z expectations + softmax-mean.
        // m = r + 8h (h = lane>=16).  q0 sign -> bit3 (=h), q1 -> (r>>2)&1,
        // q2 -> (r>>1)&1, q3 -> r&1.
#pragma unroll
        for (int grp = 0; grp < 2; ++grp) {
            const v8f& d0 = grp ? dB0 : dA0;
            const v8f& d1 = grp ? dB1 : dA1;

            float psum = 0.f, z1p = 0.f, z2p = 0.f, z3p = 0.f;
#pragma unroll
            for (int r = 0; r < 8; ++r) {
                const float p = d0[r] * d0[r] + d1[r] * d1[r];
                psum += p;
                z1p += ((r >> 2) & 1) ? -p : p;
                z2p += ((r >> 1) & 1) ? -p : p;
                z3p += (r & 1) ? -p : p;
            }
            const float z0p = lo ? psum : -psum;

            const float z0 = z0p + swap16(z0p);
            const float z1 = z1p + swap16(z1p);
            const float z2 = z2p + swap16(z2p);
            const float z3 = z3p + swap16(z3p);

            const float mz = fmaxf(fmaxf(z0, z1), fmaxf(z2, z3));
            const float e0 = __expf(z0 - mz);
            const float e1 = __expf(z1 - mz);
            const float e2 = __expf(z2 - mz);
            const float e3 = __expf(z3 - mz);
            const float inv = 1.0f / (e0 + e1 + e2 + e3);
            const float wgt = 0.25f * (e0 * inv + e1 * inv + e2 * inv + e3 * inv);

            if (lo) out[base + grp * 16 + nloc] = wgt;
        }
    }
}

extern "C" void kernel_launch(void* const* d_in, const int* in_sizes, int n_in,
                              void* d_out, int out_size, void* d_ws, size_t ws_size,
                              hipStream_t stream) {
    const float* rot = (const float*)d_in[0];    // (12,)
    const float* ent = (const float*)d_in[1];    // (3,)
    const float2* inp = (const float2*)d_in[2];  // (seq*batch) float2 angles
    float* out = (float*)d_out;
    float* V   = (float*)d_ws;                   // 32x4 floats = 512 B

    const int nsamples = in_sizes[2] / 2;
    const int nblk     = nsamples / 32;          // 32 samples per wave-iter

    qsa_precompute<<<1, 32, 0, stream>>>(rot, ent, V);

    const int threads = 256;                     // 8 waves per block
    const int wpb = threads / 32;
    int blocks = (nblk + wpb - 1) / wpb;
    if (blocks > 2048) blocks = 2048;            // grid-stride over blocks
    if (blocks < 1) blocks = 1;
    const int nwaves = blocks * wpb;

    qsa_main<<<blocks, threads, 0, stream>>>(inp, V, out, nblk, nwaves);
}